// CustomStrainEnergyLoss_84198538871524
// MI455X (gfx1250) — compile-verified
//
#include <hip/hip_runtime.h>
#include <hip/hip_bf16.h>

typedef __attribute__((ext_vector_type(2))) float v2f;
typedef __attribute__((ext_vector_type(8))) float v8f;

// ---------------------------------------------------------------------------
// zero-init d_out (harness poisons it with 0xAA)
// ---------------------------------------------------------------------------
__global__ void zero_kernel(float* out, int n) {
  int i = blockIdx.x * blockDim.x + threadIdx.x;
  if (i < n) out[i] = 0.0f;
}

// ---------------------------------------------------------------------------
// One 256-thread block per row. Streams y_pred/y_true once with coalesced
// float4 loads, applies branchless per-element trapezoid weights, then does
// the wave reduction on the matrix pipe: 64 lane-partials arranged as a
// 16x4 f32 A-matrix, multiplied by an all-ones 4x16 B via
// V_WMMA_F32_16X16X4_F32 -> 16 row sums in the D accumulator, folded with
// 7 VALU adds + one cross-half shuffle.
// ---------------------------------------------------------------------------
template <int BLOCK>
__global__ __launch_bounds__(BLOCK) void row_trapz_kernel(
    const float* __restrict__ y_pred, const float* __restrict__ y_true,
    const float* __restrict__ x, const int* __restrict__ fidx,
    float* __restrict__ row_sq,    // ws path: per-row s^2 (or nullptr)
    float* __restrict__ scalar_out,// fallback: atomic accumulate
    int N, float inv_B) {
  const int row = blockIdx.x;
  const int tid = threadIdx.x;

  int f = fidx[row];
  f = f < 0 ? 0 : (f > N - 1 ? N - 1 : f);

  const float* __restrict__ yp = y_pred + (size_t)row * N;
  const float* __restrict__ yt = y_true + (size_t)row * N;

  float p0 = 0.0f, p1 = 0.0f;

  const int stride = BLOCK * 4;        // elements covered per iteration
  const int iters  = N / stride;       // 8192 / 1024 = 8
  for (int it = 0; it < iters; ++it) {
    const int k = it * stride + tid * 4;
    const float4 a  = *(const float4*)(yp + k);
    const float4 b  = *(const float4*)(yt + k);
    const float4 xv = *(const float4*)(x + k);
    const float xm1 = x[k > 0 ? k - 1 : 0];          // guarded; unused if k==0
    const float xp4 = x[(k + 4) < N ? (k + 4) : (N - 1)];

    float d[4]  = {a.x - b.x, a.y - b.y, a.z - b.z, a.w - b.w};
    float xs[6] = {xm1, xv.x, xv.y, xv.z, xv.w, xp4};

    float acc = 0.0f;
#pragma unroll
    for (int e = 0; e < 4; ++e) {
      const int j = k + e;
      // element j weight: 0.5*(dx[j-1]*[j-1<f] + dx[j]*[j<f && j<N-1])
      const float left  = (j >= 1 && (j - 1) < f) ? (xs[e + 1] - xs[e])     : 0.0f;
      const float right = (j < N - 1 && j < f)    ? (xs[e + 2] - xs[e + 1]) : 0.0f;
      acc = __builtin_fmaf(d[e], 0.5f * (left + right), acc);
    }
    if (it & 1) p1 += acc; else p0 += acc;
  }

  // generic tail (N not a multiple of BLOCK*4) — scalar, rarely taken
  for (int j = iters * stride + tid; j < N; j += BLOCK) {
    const float d = yp[j] - yt[j];
    const float xc = x[j];
    const float left  = (j >= 1 && (j - 1) < f) ? (xc - x[j - 1]) : 0.0f;
    const float right = (j < N - 1 && j < f)    ? (x[j + 1] - xc) : 0.0f;
    p0 = __builtin_fmaf(d, 0.5f * (left + right), p0);
  }

  // ---- wave reduction via the matrix core (EXEC is all-ones here) ----
  v2f A; A[0] = p0; A[1] = p1;       // 16x4 f32 A-matrix across the wave
  v2f Bones; Bones[0] = 1.0f; Bones[1] = 1.0f;  // 4x16 all-ones B
  v8f c = {};
  c = __builtin_amdgcn_wmma_f32_16x16x4_f32(
      /*neg_a=*/false, A, /*neg_b=*/false, Bones,
      /*c_mod=*/(short)0, c, /*reuse_a=*/false, /*reuse_b=*/false);
  // lane holds rowsums {v + 8*(lane>=16)} in c[0..7]
  float t = ((c[0] + c[1]) + (c[2] + c[3])) + ((c[4] + c[5]) + (c[6] + c[7]));
  t += __shfl_xor(t, 16);            // fold the two half-wave groups

  __shared__ float wpart[BLOCK / 32];
  const int wave = tid >> 5;
  if ((tid & 31) == 0) wpart[wave] = t;
  __syncthreads();

  if (tid == 0) {
    float s = 0.0f;
#pragma unroll
    for (int w = 0; w < BLOCK / 32; ++w) s += wpart[w];
    const float sq = s * s;
    if (row_sq) row_sq[row] = sq;
    else        atomicAdd(scalar_out, sq * inv_B);
  }
}

// ---------------------------------------------------------------------------
// Mean of per-row squared errors (single small block; 16 KB of data)
// ---------------------------------------------------------------------------
__global__ __launch_bounds__(256) void mean_kernel(const float* __restrict__ row_sq,
                                                   float* __restrict__ out,
                                                   int B, float inv_B) {
  float t = 0.0f;
  for (int i = threadIdx.x; i < B; i += 256) t += row_sq[i];
#pragma unroll
  for (int o = 16; o >= 1; o >>= 1) t += __shfl_xor(t, o);
  __shared__ float wpart[8];
  if ((threadIdx.x & 31) == 0) wpart[threadIdx.x >> 5] = t;
  __syncthreads();
  if (threadIdx.x == 0) {
    float s = 0.0f;
#pragma unroll
    for (int w = 0; w < 8; ++w) s += wpart[w];
    out[0] = s * inv_B;
  }
}

// ---------------------------------------------------------------------------
extern "C" void kernel_launch(void* const* d_in, const int* in_sizes, int n_in,
                              void* d_out, int out_size, void* d_ws, size_t ws_size,
                              hipStream_t stream) {
  const float* y_pred = (const float*)d_in[0];
  const float* y_true = (const float*)d_in[1];
  const float* x      = (const float*)d_in[2];
  const int*   fidx   = (const int*)d_in[3];
  float* out = (float*)d_out;

  const int N = in_sizes[2];   // x_values length
  const int B = in_sizes[3];   // fracture_idx length (rows)
  const float inv_B = 1.0f / (float)B;

  const bool use_ws = ws_size >= (size_t)B * sizeof(float);
  float* row_sq = use_ws ? (float*)d_ws : nullptr;

  zero_kernel<<<1, 64, 0, stream>>>(out, out_size);
  row_trapz_kernel<256><<<B, 256, 0, stream>>>(y_pred, y_true, x, fidx,
                                               row_sq, out, N, inv_B);
  if (use_ws) mean_kernel<<<1, 256, 0, stream>>>(row_sq, out, B, inv_B);
}